// DecodeDetections_9869834846777
// MI455X (gfx1250) — compile-verified
//
#include <hip/hip_runtime.h>
#include <stdint.h>

#define BATCH     64
#define NBOX      65536
#define NCH       62
#define TOPK      200
#define NTHREADS  1024
#define ITERS     (NBOX / NTHREADS)

#define NBINS     8192         // pow2 scan size; key = bits >> 17 (< 0x1FC0 = 8128)
#define KEYMAX    8191u
#define KGUESS    8125u        // bits(0.9765625f) >> 17 ; speculative candidate threshold
#define CAP       2048         // candidate capacity (u64)

// smem (uint32 units), ~56.5KB, laid out disjointly so pass A can fill
// histogram AND candidates in a single streaming pass:
//   [0     .. 8191 ]  histogram -> exclusive-suffix counts (Blelloch, in place)
//   [8192  .. 12287]  cand as u64[2048]
//   [12288 .. 14335]  async-load stage (2 x 1024 dwords, double buffered)
//   [14336]           cutoff K
//   [14337]           candidate counter
#define CAND32   8192
#define STAGE32  12288
#define IDX_K    14336
#define IDX_CNT  14337
#define SMEMSZ   14464

__device__ __forceinline__ void async_gather_conf(uint32_t stage_base_lds, int it, int buf,
                                                  int tid, uint64_t gconf)
{
    uint32_t box  = (uint32_t)(it * NTHREADS + tid);
    uint32_t voff = box * (uint32_t)(NCH * 4);                       // byte offset of conf[box]
    uint32_t dst  = stage_base_lds + ((uint32_t)buf << 12) + (uint32_t)tid * 4u;
    asm volatile("global_load_async_to_lds_b32 %0, %1, %2"
                 :: "v"(dst), "v"(voff), "s"(gconf) : "memory");
}

__global__ __launch_bounds__(NTHREADS)
void decode_detections_topk(const float* __restrict__ y, float* __restrict__ out)
{
    __shared__ __attribute__((aligned(16))) uint32_t smem[SMEMSZ];
    const int tid = threadIdx.x;
    const int b   = blockIdx.x;
    const float* yb = y + (size_t)b * NBOX * NCH;

    for (int i = tid; i < SMEMSZ; i += NTHREADS) smem[i] = 0u;
    __syncthreads();

    unsigned long long* cand = (unsigned long long*)&smem[CAND32];
    volatile uint32_t*  stage = &smem[STAGE32];
    const uint64_t gconf = (uint64_t)(uintptr_t)(yb + 1);            // conf channel base
    const uint32_t stage_base_lds = (uint32_t)(uintptr_t)&smem[STAGE32];

    // ---------- Pass A (single streaming pass):
    //   async-gather conf -> LDS (double buffered), histogram keys,
    //   and speculatively collect candidates with key >= KGUESS. ----------
    async_gather_conf(stage_base_lds, 0, 0, tid, gconf);
    for (int it = 0; it < ITERS; ++it) {
        if (it + 1 < ITERS) {
            if (it + 2 < ITERS)   // warm L2 two iterations ahead (global_prefetch_b8)
                __builtin_prefetch(yb + (size_t)((it + 2) * NTHREADS + tid) * NCH + 1, 0, 0);
            async_gather_conf(stage_base_lds, it + 1, (it + 1) & 1, tid, gconf);
            asm volatile("s_wait_asynccnt 0x1" ::: "memory");        // oldest gather done
        } else {
            asm volatile("s_wait_asynccnt 0x0" ::: "memory");
        }
        uint32_t box  = (uint32_t)(it * NTHREADS + tid);
        float c = __uint_as_float(stage[(it & 1) * NTHREADS + tid]);
        c = (c >= 0.01f) ? c : 0.0f;
        uint32_t bits = __float_as_uint(c);
        uint32_t key  = bits >> 17;
        if (key > KEYMAX) key = KEYMAX;
        atomicAdd(&smem[key], 1u);
        if (key >= KGUESS) {
            uint32_t pos = atomicAdd(&smem[IDX_CNT], 1u);
            if (pos < CAP)
                cand[pos] = ((unsigned long long)bits << 32) | (uint32_t)(~box);
        }
    }
    __syncthreads();

    // ---------- In-place Blelloch exclusive-SUFFIX scan over NBINS ----------
    // Operate on reversed index: A(r) == smem[NBINS-1-r].
    for (int d = 1; d < NBINS; d <<= 1) {
        for (int p = tid; p < NBINS / (2 * d); p += NTHREADS) {
            int r1 = 2 * d * p + d - 1;
            int r2 = 2 * d * p + 2 * d - 1;
            smem[NBINS - 1 - r2] += smem[NBINS - 1 - r1];
        }
        __syncthreads();
    }
    if (tid == 0) smem[0] = 0u;          // A(NBINS-1) = 0 (root held the total)
    __syncthreads();
    for (int d = NBINS >> 1; d >= 1; d >>= 1) {
        for (int p = tid; p < NBINS / (2 * d); p += NTHREADS) {
            int r1 = 2 * d * p + d - 1;
            int r2 = 2 * d * p + 2 * d - 1;
            uint32_t t = smem[NBINS - 1 - r1];
            smem[NBINS - 1 - r1] = smem[NBINS - 1 - r2];
            smem[NBINS - 1 - r2] += t;
        }
        __syncthreads();
    }
    // now smem[k] = count of values with key > k

    // ---------- Cutoff K: smem[K] < TOPK <= (K ? smem[K-1] : NBOX) ----------
    for (int K = tid; K <= (int)KEYMAX; K += NTHREADS) {
        uint32_t e  = smem[K];
        uint32_t e1 = (K == 0) ? (uint32_t)NBOX : smem[K - 1];
        if (e < TOPK && e1 >= TOPK) smem[IDX_K] = (uint32_t)K;
    }
    __syncthreads();
    const uint32_t Kcut = smem[IDX_K];
    const uint32_t cnt  = smem[IDX_CNT];
    __syncthreads();

    // Fast path valid iff speculative set is a superset of {key >= Kcut} and nothing
    // was dropped. (Block-uniform condition -> uniform branch around barriers.)
    if (!(Kcut >= KGUESS && cnt <= CAP)) {
        // ---------- Fallback pass B: exact re-scan with cutoff Kcut ----------
        for (int i = CAND32; i < CAND32 + 2 * CAP; i += 1) { }       // (no-op; kept simple below)
        for (int i = tid; i < 2 * CAP; i += NTHREADS) smem[CAND32 + i] = 0u;
        if (tid == 0) smem[IDX_CNT] = 0u;
        __syncthreads();
        for (int it = 0; it < ITERS; ++it) {
            async_gather_conf(stage_base_lds, it, 0, tid, gconf);
            asm volatile("s_wait_asynccnt 0x0" ::: "memory");
            uint32_t box = (uint32_t)(it * NTHREADS + tid);
            float c = __uint_as_float(stage[tid]);
            c = (c >= 0.01f) ? c : 0.0f;
            uint32_t bits = __float_as_uint(c);
            uint32_t key  = bits >> 17;
            if (key > KEYMAX) key = KEYMAX;
            if (key >= Kcut) {
                uint32_t pos = atomicAdd(&smem[IDX_CNT], 1u);
                if (pos < CAP)
                    cand[pos] = ((unsigned long long)bits << 32) | (uint32_t)(~box);
            }
            asm volatile("s_wait_dscnt 0x0" ::: "memory");           // stage consumed before reuse
        }
        __syncthreads();
    }

    // ---------- Bitonic sort: 2048 x u64, DESCENDING (value desc, index asc) ----------
    for (unsigned k = 2; k <= CAP; k <<= 1) {
        for (unsigned j = k >> 1; j >= 1; j >>= 1) {
            for (unsigned i = tid; i < CAP; i += NTHREADS) {
                unsigned l = i ^ j;
                if (l > i) {
                    unsigned long long a = cand[i], bb = cand[l];
                    bool up = ((i & k) == 0);
                    bool sw = up ? (a < bb) : (a > bb);
                    if (sw) { cand[i] = bb; cand[l] = a; }
                }
            }
            __syncthreads();
        }
    }

    // ---------- Decode top-200 and write (b, t, 0..52) ----------
    for (int jj = tid; jj < TOPK * 53; jj += NTHREADS) {
        int t = jj / 53, c = jj % 53;
        unsigned long long pk = cand[t];
        uint32_t bits = (uint32_t)(pk >> 32);
        uint32_t idx  = (~(uint32_t)pk) & 0xFFFFu;
        float* o = out + ((size_t)b * TOPK + t) * 53 + c;
        if (c == 0) {
            *o = __uint_as_float(bits);
        } else {
            int cc  = c - 1;
            int par = cc & 1;
            const float* yr = yb + (size_t)idx * NCH;
            float v = yr[2 + cc] * yr[56 + par] * yr[58 + par] + yr[54 + par];
            *o = v * 512.0f;
        }
    }
}

extern "C" void kernel_launch(void* const* d_in, const int* in_sizes, int n_in,
                              void* d_out, int out_size, void* d_ws, size_t ws_size,
                              hipStream_t stream) {
    (void)in_sizes; (void)n_in; (void)d_ws; (void)ws_size; (void)out_size;
    const float* y = (const float*)d_in[0];
    float* out = (float*)d_out;
    hipLaunchKernelGGL(decode_detections_topk, dim3(BATCH), dim3(NTHREADS), 0, stream,
                       y, out);
}